// ExtractorMLP_28192165331244
// MI455X (gfx1250) — compile-verified
//
#include <hip/hip_runtime.h>
#include <math.h>

typedef float v2f __attribute__((ext_vector_type(2)));
typedef float v8f __attribute__((ext_vector_type(8)));

#define H    128
#define EPSN 1e-8f

// ---------------------------------------------------------------------------
// Kernel 1: per-node MLP  Y = (ELU(emb @ W1^T + b1)) @ W2^T + b2, plus row
// norms. One wave32 per 16-node row block, fp32 WMMA (v_wmma_f32_16x16x4_f32).
// A-frag layout (ISA 7.12.2, 32-bit A 16x4): lane L holds row M=L%16,
//   VGPR0 = K+(L<16?0:2), VGPR1 = K+(L<16?1:3).
// B = W^T slice (4x16): lane L holds col N=L%16 with same K split -> a
//   contiguous float2 from W row N (W[n][k], W[n][k+1]).
// D layout: VGPR r -> rows r (lanes 0-15) and r+8 (lanes 16-31), col = lane%16.
// ---------------------------------------------------------------------------
__global__ __launch_bounds__(32)
void node_mlp_kernel(const float* __restrict__ emb,
                     const float* __restrict__ W1, const float* __restrict__ b1,
                     const float* __restrict__ W2, const float* __restrict__ b2,
                     float* __restrict__ Y, float* __restrict__ nrm, int N)
{
    __shared__ float lds[16 * H];        // staging: X -> h -> y (8 KB)
    const int lane = threadIdx.x;        // 0..31
    const int m    = lane & 15;          // row (A) / col (B) within tile
    const int kh   = (lane >> 4) * 2;    // K sub-offset per half-wave
    const int rhalf = (lane >> 4) << 3;  // 0 or 8 (D-layout row offset)
    const int r0   = blockIdx.x * 16;

    // ---- stage X tile [16][128] into LDS (coalesced) ----
    for (int i = lane; i < 16 * H; i += 32) {
        int r = r0 + (i >> 7);
        lds[i] = (r < N) ? emb[(size_t)r * H + (i & 127)] : 0.0f;
    }
    __syncthreads();

    v8f acc[8];
    const v8f vzero = {0.f, 0.f, 0.f, 0.f, 0.f, 0.f, 0.f, 0.f};

    // ---- layer 1: h = ELU(X @ W1^T + b1) ----
    #pragma unroll
    for (int n = 0; n < 8; ++n) acc[n] = vzero;
    for (int k = 0; k < H; k += 4) {
        v2f a = { lds[m * H + k + kh], lds[m * H + k + kh + 1] };
        #pragma unroll
        for (int n = 0; n < 8; ++n) {
            const float* wrow = W1 + (size_t)(n * 16 + m) * H + k + kh;
            v2f b = { wrow[0], wrow[1] };
            acc[n] = __builtin_amdgcn_wmma_f32_16x16x4_f32(
                false, a, false, b, (short)0, acc[n], false, false);
        }
    }
    __syncthreads();
    // D-layout -> row-major h in LDS, with bias + ELU
    #pragma unroll
    for (int n = 0; n < 8; ++n) {
        int col = n * 16 + m;
        float bias = b1[col];
        #pragma unroll
        for (int r = 0; r < 8; ++r) {
            float x = acc[n][r] + bias;
            lds[(r + rhalf) * H + col] = (x > 0.0f) ? x : expm1f(x);
        }
    }
    __syncthreads();

    // ---- layer 2: y = h @ W2^T + b2 ----
    #pragma unroll
    for (int n = 0; n < 8; ++n) acc[n] = vzero;
    for (int k = 0; k < H; k += 4) {
        v2f a = { lds[m * H + k + kh], lds[m * H + k + kh + 1] };
        #pragma unroll
        for (int n = 0; n < 8; ++n) {
            const float* wrow = W2 + (size_t)(n * 16 + m) * H + k + kh;
            v2f b = { wrow[0], wrow[1] };
            acc[n] = __builtin_amdgcn_wmma_f32_16x16x4_f32(
                false, a, false, b, (short)0, acc[n], false, false);
        }
    }
    __syncthreads();
    #pragma unroll
    for (int n = 0; n < 8; ++n) {
        int col = n * 16 + m;
        float bias = b2[col];
        #pragma unroll
        for (int r = 0; r < 8; ++r)
            lds[(r + rhalf) * H + col] = acc[n][r] + bias;
    }
    __syncthreads();

    // ---- per-row L2 norms (lanes 0..15, one row each) ----
    if (lane < 16) {
        int r = r0 + lane;
        if (r < N) {
            float s = 0.0f;
            #pragma unroll 4
            for (int j = 0; j < H; ++j) {
                float v = lds[lane * H + j];
                s += v * v;
            }
            nrm[r] = fmaxf(sqrtf(s), EPSN);
        }
    }
    // ---- store Y (coalesced) ----
    for (int i = lane; i < 16 * H; i += 32) {
        int r = r0 + (i >> 7);
        if (r < N) Y[(size_t)r * H + (i & 127)] = lds[i];
    }
}

// ---------------------------------------------------------------------------
// Kernel 2: one wave32 per edge. Each lane takes a float4 slice of both
// endpoint feature rows (coalesced 512B row reads, L2-resident), wave-reduce
// the dot product, lane 0 writes cos = dot / (n1*n2).
// ---------------------------------------------------------------------------
__global__ __launch_bounds__(256)
void edge_cos_kernel(const int* __restrict__ ei, const float* __restrict__ Y,
                     const float* __restrict__ nrm, float* __restrict__ out,
                     int E)
{
    int wid  = (int)((blockIdx.x * 256u + threadIdx.x) >> 5);
    int lane = threadIdx.x & 31;
    if (wid >= E) return;
    int c = ei[wid];         // edge_index[0][e]
    int r = ei[E + wid];     // edge_index[1][e]
    const float4* f1 = (const float4*)(Y + (size_t)c * H);
    const float4* f2 = (const float4*)(Y + (size_t)r * H);
    float4 a = f1[lane];
    float4 b = f2[lane];
    float s = a.x * b.x + a.y * b.y + a.z * b.z + a.w * b.w;
    #pragma unroll
    for (int off = 16; off > 0; off >>= 1)
        s += __shfl_xor(s, off, 32);
    if (lane == 0) out[wid] = s / (nrm[c] * nrm[r]);
}

// ---------------------------------------------------------------------------
extern "C" void kernel_launch(void* const* d_in, const int* in_sizes, int n_in,
                              void* d_out, int out_size, void* d_ws, size_t ws_size,
                              hipStream_t stream)
{
    const float* emb = (const float*)d_in[0];
    const int*   ei  = (const int*)  d_in[1];   // edge_index, [2, E] flat
    const float* W1  = (const float*)d_in[2];
    const float* b1  = (const float*)d_in[3];
    const float* W2  = (const float*)d_in[4];
    const float* b2  = (const float*)d_in[5];
    float* out = (float*)d_out;

    const int N = in_sizes[0] / H;   // 50000
    const int E = in_sizes[1] / 2;   // 800000

    // Workspace: Y [N*H] f32 (25.6 MB) then nrm [N] f32 (0.2 MB)
    float* Y   = (float*)d_ws;
    float* nv  = Y + (size_t)N * H;

    int nblk = (N + 15) / 16;
    hipLaunchKernelGGL(node_mlp_kernel, dim3(nblk), dim3(32), 0, stream,
                       emb, W1, b1, W2, b2, Y, nv, N);

    int eblk = (E + 7) / 8;          // 8 waves (8 edges) per 256-thread block
    hipLaunchKernelGGL(edge_cos_kernel, dim3(eblk), dim3(256), 0, stream,
                       ei, Y, nv, out, E);
}